// MDCG_6270652252524
// MI455X (gfx1250) — compile-verified
//
#include <hip/hip_runtime.h>

typedef __attribute__((ext_vector_type(2))) float v2f;
typedef __attribute__((ext_vector_type(8))) float v8f;

#define NNODES 100000
#define DFEAT  128
#define NEDGES 1600000
#define TAU_INV 2.0f            // 1/0.5
#define LAYER_DROPOUT 0.5f

// ---------------------------------------------------------------------------
// CDNA5 async copy global->LDS (ASYNCcnt path, CDNA5 ISA).
// GV mode: VDST = per-lane LDS byte address, VADDR = 64-bit global address.
// ---------------------------------------------------------------------------
__device__ __forceinline__ uint32_t lds_addr_of(const void* p) {
    // LDS aperture: ADDR[31:0] is the raw LDS byte offset
    return (uint32_t)(uintptr_t)p;
}
__device__ __forceinline__ void async_copy_b128(uint32_t lds_addr, const void* gaddr) {
    asm volatile("global_load_async_to_lds_b128 %0, %1, off"
                 :: "v"(lds_addr), "v"((unsigned long long)(uintptr_t)gaddr)
                 : "memory");
}
__device__ __forceinline__ void wait_async0() {
    asm volatile("s_wait_asynccnt 0" ::: "memory");
}

// ---------------------------------------------------------------------------
// Kernel 1: support = x @ W   via V_WMMA_F32_16X16X4_F32, operands LDS-staged.
// Block = 256 thr = 8 waves; tileM = blockIdx.x, tileN = waveId.
// W (64KB) + x slab (8KB) async-copied to LDS once; inner loop is ds_load+wmma.
// ---------------------------------------------------------------------------
__global__ __launch_bounds__(256) void gemm_support(const float* __restrict__ x,
                                                    const float* __restrict__ w,
                                                    float* __restrict__ support) {
    __shared__ float sW[DFEAT * DFEAT];   // 64 KB
    __shared__ float sX[16 * DFEAT];      // 8 KB
    const int tid  = threadIdx.x;
    const int wave = tid >> 5;
    const int lane = tid & 31;
    const int half = lane >> 4;
    const int l16  = lane & 15;
    const int m0 = blockIdx.x * 16;
    const int n0 = wave * 16;

    // stage W: 4096 float4, 16 per thread (async -> LDS)
    const uint32_t swb = lds_addr_of(sW);
    #pragma unroll
    for (int i = 0; i < 16; ++i) {
        const int f = tid + i * 256;
        async_copy_b128(swb + (uint32_t)f * 16, (const char*)w + (size_t)f * 16);
    }
    // stage x slab: 512 float4, 2 per thread
    const uint32_t sxb = lds_addr_of(sX);
    const char* xslab = (const char*)(x + (size_t)m0 * DFEAT);
    #pragma unroll
    for (int i = 0; i < 2; ++i) {
        const int f = tid + i * 256;
        async_copy_b128(sxb + (uint32_t)f * 16, xslab + (size_t)f * 16);
    }
    wait_async0();
    __syncthreads();

    v8f acc = {};
    #pragma unroll 4
    for (int ks = 0; ks < DFEAT / 4; ++ks) {
        const int kk = ks * 4 + 2 * half;
        v2f a, b;
        a.x = sX[l16 * DFEAT + kk];
        a.y = sX[l16 * DFEAT + kk + 1];
        b.x = sW[kk * DFEAT + n0 + l16];
        b.y = sW[(kk + 1) * DFEAT + n0 + l16];
        acc = __builtin_amdgcn_wmma_f32_16x16x4_f32(false, a, false, b,
                                                    (short)0, acc, false, false);
    }
    #pragma unroll
    for (int v = 0; v < 8; ++v) {
        const int row = m0 + v + 8 * half;
        support[(size_t)row * DFEAT + n0 + l16] = acc[v];
    }
}

// ---------------------------------------------------------------------------
// Kernel 2: scatter-add SpMM.  One wave per edge; each lane moves a float4 and
// issues 4 hardware fp32 atomics (51MB target is L2-resident -> L2 atomic units)
// ---------------------------------------------------------------------------
__global__ __launch_bounds__(256) void spmm_scatter(const int* __restrict__ edge_row,
                                                    const int* __restrict__ edge_col,
                                                    const float* __restrict__ edge_val,
                                                    const float* __restrict__ support,
                                                    float* __restrict__ agg) {
    const int e    = blockIdx.x * 8 + (threadIdx.x >> 5);
    const int lane = threadIdx.x & 31;
    const int src = edge_col[e];
    const int dst = edge_row[e];
    const float val = edge_val[e];

    const float4 m = ((const float4*)(support + (size_t)src * DFEAT))[lane];
    float* drow = agg + (size_t)dst * DFEAT + lane * 4;
    unsafeAtomicAdd(drow + 0, val * m.x);
    unsafeAtomicAdd(drow + 1, val * m.y);
    unsafeAtomicAdd(drow + 2, val * m.z);
    unsafeAtomicAdd(drow + 3, val * m.w);
}

// ---------------------------------------------------------------------------
// Kernel 3: h = relu(agg + bias) stored back, out = x + 0.5 * h
// ---------------------------------------------------------------------------
__global__ __launch_bounds__(256) void bias_relu_out(const float* __restrict__ x,
                                                     const float* __restrict__ bias,
                                                     float* __restrict__ hbuf,
                                                     float* __restrict__ out) {
    const size_t i = (size_t)blockIdx.x * 256 + threadIdx.x;
    float h = hbuf[i] + bias[i & (DFEAT - 1)];
    h = fmaxf(h, 0.0f);
    hbuf[i] = h;
    out[i] = x[i] + LAYER_DROPOUT * h;
}

// ---------------------------------------------------------------------------
// Kernel 4: gate MLP + gumbel softmax.
// Per wave: one 16-row slab. A = [x, x+h] slab (16x256, 16KB) built once in LDS
// per wave; the wc0 n-tile (256x16, 16KB) is async-staged per nt and shared by
// all 8 waves.  Out-of-range waves clamp (no early return -> barriers safe,
// EXEC all-1s for WMMA) and skip the final store.
// ---------------------------------------------------------------------------
__global__ __launch_bounds__(256) void gate_kernel(const float* __restrict__ x,
                                                   const float* __restrict__ hbuf,
                                                   const float* __restrict__ wc0,
                                                   const float* __restrict__ wc1,
                                                   const float* __restrict__ node_lastlayer,
                                                   const float* __restrict__ gnoise,
                                                   float* __restrict__ node_choose) {
    __shared__ float sA[8][16 * 256];    // 128 KB : per-wave A slab
    __shared__ float sB[256 * 16];       // 16 KB  : wc0 n-tile (k-major, 16 cols)

    const int wave = threadIdx.x >> 5;
    const int lane = threadIdx.x & 31;
    const int half = lane >> 4;
    const int l16  = lane & 15;
    const int waveG = blockIdx.x * 8 + wave;
    int m0 = waveG * 16;
    const bool active = (m0 < NNODES);
    if (!active) m0 = NNODES - 16;       // clamp: redundant compute, no store

    // ---- stage A slab for this wave: rows m0..m0+15, k 0..255 ----
    float* sAw = &sA[wave][0];
    const float* xs = x    + (size_t)m0 * DFEAT;
    const float* hs = hbuf + (size_t)m0 * DFEAT;
    #pragma unroll 4
    for (int it = 0; it < 32; ++it) {
        const int f = lane + it * 32;    // float4 index in 16x256 slab (1024 total)
        const int r = f >> 6;            // row (64 float4 per row)
        const int q = f & 63;            // float4 within row
        float4 v;
        if (q < 32) {                    // k < 128 : x
            v = ((const float4*)(xs + (size_t)r * DFEAT))[q];
        } else {                         // k >= 128 : x + h
            const float4 xv = ((const float4*)(xs + (size_t)r * DFEAT))[q - 32];
            const float4 hv = ((const float4*)(hs + (size_t)r * DFEAT))[q - 32];
            v = make_float4(xv.x + hv.x, xv.y + hv.y, xv.z + hv.z, xv.w + hv.w);
        }
        ((float4*)sAw)[f] = v;
    }

    const uint32_t sbb = lds_addr_of(sB);
    float s0[8] = {0.f, 0.f, 0.f, 0.f, 0.f, 0.f, 0.f, 0.f};
    float s1[8] = {0.f, 0.f, 0.f, 0.f, 0.f, 0.f, 0.f, 0.f};

    for (int nt = 0; nt < 16; ++nt) {
        __syncthreads();                 // previous sB tile fully consumed
        const int nbase = nt * 16;
        // stage wc0[:, nbase..nbase+15] -> sB[k*16 + c]  (1024 float4, 4/thread)
        #pragma unroll
        for (int i = 0; i < 4; ++i) {
            const int f  = threadIdx.x + i * 256;
            const int k  = f >> 2;
            const int c4 = f & 3;
            async_copy_b128(sbb + (uint32_t)f * 16,
                            wc0 + (size_t)k * (2 * DFEAT) + nbase + c4 * 4);
        }
        wait_async0();
        __syncthreads();

        v8f acc = {};
        #pragma unroll 4
        for (int ks = 0; ks < 64; ++ks) {
            const int kk = ks * 4 + 2 * half;
            v2f a, b;
            a.x = sAw[l16 * 256 + kk];
            a.y = sAw[l16 * 256 + kk + 1];
            b.x = sB[kk * 16 + l16];
            b.y = sB[(kk + 1) * 16 + l16];
            acc = __builtin_amdgcn_wmma_f32_16x16x4_f32(false, a, false, b,
                                                        (short)0, acc, false, false);
        }
        const int n = nbase + l16;
        const float w0 = wc1[(size_t)n * 2 + 0];
        const float w1 = wc1[(size_t)n * 2 + 1];
        #pragma unroll
        for (int v = 0; v < 8; ++v) {
            const float t = fmaxf(acc[v], 0.0f);   // relu
            s0[v] += t * w0;
            s1[v] += t * w1;
        }
    }

    // reduce over the 16 lanes of each half (xor over lane bits 0..3)
    #pragma unroll
    for (int v = 0; v < 8; ++v) {
        #pragma unroll
        for (int mask = 1; mask < 16; mask <<= 1) {
            s0[v] += __shfl_xor(s0[v], mask, 32);
            s1[v] += __shfl_xor(s1[v], mask, 32);
        }
    }

    // lanes 0 and 16 finalize 8 rows each: gumbel softmax + straight-through
    if (active && l16 == 0) {
        #pragma unroll
        for (int v = 0; v < 8; ++v) {
            const int r = m0 + half * 8 + v;
            const float l0 = (s0[v] + gnoise[(size_t)r * 2 + 0]) * TAU_INV;
            const float l1 = (s1[v] + gnoise[(size_t)r * 2 + 1]) * TAU_INV;
            const float mx = fmaxf(l0, l1);
            const float e0 = __expf(l0 - mx);
            const float e1 = __expf(l1 - mx);
            const float ysoft0 = e0 / (e0 + e1);
            const float hard0 = (l1 > l0) ? 0.0f : 1.0f;
            const float y0 = hard0 - ysoft0 + ysoft0;   // straight-through value
            node_choose[r] = node_lastlayer[r] * y0;
        }
    }
}

// ---------------------------------------------------------------------------
extern "C" void kernel_launch(void* const* d_in, const int* in_sizes, int n_in,
                              void* d_out, int out_size, void* d_ws, size_t ws_size,
                              hipStream_t stream) {
    (void)in_sizes; (void)n_in; (void)out_size; (void)ws_size;

    const float* x    = (const float*)d_in[0];
    const int*   er   = (const int*)d_in[1];
    const int*   ec   = (const int*)d_in[2];
    const float* ev   = (const float*)d_in[3];
    const float* w    = (const float*)d_in[4];
    const float* bias = (const float*)d_in[5];
    const float* wc0  = (const float*)d_in[6];
    const float* wc1  = (const float*)d_in[7];
    const float* nll  = (const float*)d_in[8];
    const float* gn   = (const float*)d_in[9];

    float* out         = (float*)d_out;                       // [N,128]
    float* node_choose = out + (size_t)NNODES * DFEAT;        // [N,1]

    float* support = (float*)d_ws;                            // [N,128] 51.2 MB
    float* hbuf    = support + (size_t)NNODES * DFEAT;        // [N,128] 51.2 MB

    hipMemsetAsync(hbuf, 0, (size_t)NNODES * DFEAT * sizeof(float), stream);

    gemm_support<<<NNODES / 16, 256, 0, stream>>>(x, w, support);
    spmm_scatter<<<NEDGES / 8, 256, 0, stream>>>(er, ec, ev, support, hbuf);
    bias_relu_out<<<(NNODES * DFEAT) / 256, 256, 0, stream>>>(x, bias, hbuf, out);
    gate_kernel<<<(NNODES / 16 + 7) / 8, 256, 0, stream>>>(x, hbuf, wc0, wc1, nll, gn,
                                                           node_choose);
}